// BoundaryLoss_16509854286366
// MI455X (gfx1250) — compile-verified
//
#include <hip/hip_runtime.h>
#include <stdint.h>

#define BATCH 8
#define NCLS  21
#define HH    512
#define WW    512
#define PLANE (HH * WW)                 // 262144
#define NPIX  (BATCH * HH * WW)         // 2097152
#define IGNORE_INDEX 255
#define NBLOCKS_CE (BATCH * HH)         // 4096 partials

// ---- CDNA5 feature detection (compile-safe fallbacks everywhere) ----
#if defined(__has_builtin)
#  if __has_builtin(__builtin_amdgcn_global_load_async_to_lds_b32)
#    define USE_ASYNC_LDS 1
#  else
#    define USE_ASYNC_LDS 0
#  endif
#  if __has_builtin(__builtin_amdgcn_s_wait_asynccnt)
#    define HAVE_WAIT_ASYNC 1
#  else
#    define HAVE_WAIT_ASYNC 0
#  endif
#  if __has_builtin(__builtin_amdgcn_wmma_f32_16x16x4_f32)
#    define USE_WMMA_F32X4 1
#  else
#    define USE_WMMA_F32X4 0
#  endif
#else
#  define USE_ASYNC_LDS 0
#  define HAVE_WAIT_ASYNC 0
#  define USE_WMMA_F32X4 0
#endif

typedef __attribute__((ext_vector_type(2))) float v2f;
typedef __attribute__((ext_vector_type(8))) float v8f;

typedef __attribute__((address_space(1))) int as1_int;
typedef __attribute__((address_space(3))) int as3_int;

__device__ __forceinline__ void wait_async_zero() {
#if HAVE_WAIT_ASYNC
  __builtin_amdgcn_s_wait_asynccnt(0);
#else
  asm volatile("s_wait_asynccnt 0" ::: "memory");
#endif
}

// ---------------------------------------------------------------------
// Pass 1: boundary map from targets.
// Faithful semantics: vdiff(b,i,j) = (max over rows i-1..i+1 at col j) !=
// (min over same); bmap(i,j) = OR over b and cols j-1..j+1 of vdiff.
// Tile of targets staged into LDS via async global->LDS loads.
// Block: (64,4) threads -> 64x4 output tile; LDS tile 8 x 6 x 66 ints.
// ---------------------------------------------------------------------
__global__ __launch_bounds__(256) void boundary_kernel(
    const int* __restrict__ tgt, unsigned char* __restrict__ bmap) {
  __shared__ int sm[BATCH][6][66];

  const int tx = threadIdx.x;            // 0..63
  const int ty = threadIdx.y;            // 0..3
  const int tid = ty * 64 + tx;
  const int j0 = blockIdx.x * 64;
  const int i0 = blockIdx.y * 4;

  // Load tile rows [i0-1, i0+4], cols [j0-1, j0+64], clamped at image edges
  // (clamped cells only feed edge outputs, which are forced to zero anyway).
  const int NELEM = BATCH * 6 * 66;      // 3168
  for (int e = tid; e < NELEM; e += 256) {
    int c = e % 66;
    int r = (e / 66) % 6;
    int b = e / (66 * 6);
    int gi = min(max(i0 - 1 + r, 0), HH - 1);
    int gj = min(max(j0 - 1 + c, 0), WW - 1);
    const int* gp = tgt + ((size_t)b * HH + gi) * WW + gj;
#if USE_ASYNC_LDS
    __builtin_amdgcn_global_load_async_to_lds_b32(
        (as1_int*)(uintptr_t)gp,
        (as3_int*)(uint32_t)(uintptr_t)&sm[b][r][c],
        0, 0);
#else
    sm[b][r][c] = *gp;
#endif
  }
#if USE_ASYNC_LDS
  wait_async_zero();
#endif
  __syncthreads();

  const int i = i0 + ty;
  const int j = j0 + tx;
  unsigned char out = 0;
  if (i >= 1 && i <= HH - 2 && j >= 1 && j <= WW - 2) {
    bool any = false;
#pragma unroll
    for (int b = 0; b < BATCH; ++b) {
#pragma unroll
      for (int dc = 0; dc < 3; ++dc) {
        int a0 = sm[b][ty + 0][tx + dc];
        int a1 = sm[b][ty + 1][tx + dc];
        int a2 = sm[b][ty + 2][tx + dc];
        int mx = max(a0, max(a1, a2));
        int mn = min(a0, min(a1, a2));
        any = any || (mx != mn);
      }
    }
    out = any ? (unsigned char)1 : (unsigned char)0;
  }
  bmap[i * WW + j] = out;
}

// ---------------------------------------------------------------------
// Pass 2: per-pixel weighted cross entropy + deterministic block reduce.
// One block per (b, row); 512 threads = one pixel each. 21 plane reads
// per thread are coalesced across the row (consecutive j -> consecutive
// addresses within each channel plane).
// ---------------------------------------------------------------------
__global__ __launch_bounds__(512) void ce_kernel(
    const float* __restrict__ x, const int* __restrict__ tgt,
    const unsigned char* __restrict__ bmap, float* __restrict__ partials) {
  const int tid = threadIdx.x;
  const int bi  = blockIdx.x;            // 0..4095
  const int b   = bi >> 9;
  const int i   = bi & (HH - 1);
  const int j   = tid;

  const float* px = x + (((size_t)b * NCLS) * HH + i) * WW + j;
  const int t = tgt[((size_t)b * HH + i) * WW + j];

  float v[NCLS];
  float m = -3.402823466e38f;
#pragma unroll
  for (int c = 0; c < NCLS; ++c) {
    v[c] = px[(size_t)c * PLANE];
    m = fmaxf(m, v[c]);
  }
  float s = 0.0f, xt = 0.0f;
#pragma unroll
  for (int c = 0; c < NCLS; ++c) {
    s += __expf(v[c] - m);
    xt = (t == c) ? v[c] : xt;
  }
  float ce = (t != IGNORE_INDEX) ? (m + __logf(s) - xt) : 0.0f;
  float w  = 1.0f + 2.0f * (float)bmap[i * WW + j];
  float val = ce * w;

  __shared__ float red[512];
  red[tid] = val;
  __syncthreads();
#pragma unroll
  for (int off = 256; off > 0; off >>= 1) {
    if (tid < off) red[tid] += red[tid + off];
    __syncthreads();
  }
  if (tid == 0) partials[bi] = red[0];
}

// ---------------------------------------------------------------------
// Pass 3: sum 4096 partials with V_WMMA_F32_16X16X4_F32 (exact fp32 FMA
// chain, fixed order -> deterministic). A = 64 partials per step laid out
// per the 16x4 f32 A layout (lanes 0-15 hold K=0,1; lanes 16-31 hold
// K=2,3); B = all ones, so D[m,n] = sum_k A[m,k]. Accumulate C across 64
// steps. D layout: VGPR v at lane 0 = (M=v, N=0), lane 16 = (M=v+8, N=0),
// so total = (sum of 8 acc comps)@lane0 + @lane16.
// Single wave of 32 threads, no divergence before the WMMA (EXEC all 1s).
// ---------------------------------------------------------------------
__global__ __launch_bounds__(32) void finalize_kernel(
    const float* __restrict__ partials, float* __restrict__ out) {
  const int lane = threadIdx.x;
#if USE_WMMA_F32X4
  const int m = lane & 15;
  const int o = (lane >= 16) ? 2 : 0;
  v8f acc = {0.f, 0.f, 0.f, 0.f, 0.f, 0.f, 0.f, 0.f};
  v2f ones; ones.x = 1.0f; ones.y = 1.0f;
  for (int k = 0; k < NBLOCKS_CE / 64; ++k) {
    const float* p = partials + k * 64 + m * 4 + o;
    v2f a; a.x = p[0]; a.y = p[1];
    acc = __builtin_amdgcn_wmma_f32_16x16x4_f32(
        false, a, false, ones, (short)0, acc, false, false);
  }
  float sv = acc[0] + acc[1] + acc[2] + acc[3] + acc[4] + acc[5] + acc[6] + acc[7];
  float total = __shfl(sv, 0, 32) + __shfl(sv, 16, 32);
  if (lane == 0) out[0] = total / (float)NPIX;
#else
  float s = 0.0f;
  for (int k = lane; k < NBLOCKS_CE; k += 32) s += partials[k];
#pragma unroll
  for (int off = 16; off > 0; off >>= 1) s += __shfl_down(s, off, 32);
  if (lane == 0) out[0] = s / (float)NPIX;
#endif
}

extern "C" void kernel_launch(void* const* d_in, const int* in_sizes, int n_in,
                              void* d_out, int out_size, void* d_ws, size_t ws_size,
                              hipStream_t stream) {
  const float* logits  = (const float*)d_in[0];       // (8,21,512,512) fp32
  const int*   targets = (const int*)d_in[1];         // (8,512,512) int
  unsigned char* bmap  = (unsigned char*)d_ws;        // 512*512 bytes
  float* partials = (float*)((char*)d_ws + (size_t)HH * WW);  // 4096 floats
  float* out = (float*)d_out;

  dim3 bblock(64, 4);
  dim3 bgrid(WW / 64, HH / 4);
  boundary_kernel<<<bgrid, bblock, 0, stream>>>(targets, bmap);
  ce_kernel<<<NBLOCKS_CE, 512, 0, stream>>>(logits, targets, bmap, partials);
  finalize_kernel<<<1, 32, 0, stream>>>(partials, out);
}